// MultiTapDelay_73160472920636
// MI455X (gfx1250) — compile-verified
//
#include <hip/hip_runtime.h>
#include <stdint.h>

// ---------------------------------------------------------------------------
// MultiTapDelay on MI455X (gfx1250)
// out[t] = 0.5*x[t] + 0.5 * sum_i g_i * x[t - s_i]   (zero for t < s_i)
// Memory-bound: ~212 MB min traffic -> ~9.1 us @ 23.3 TB/s. Strategy: TDM
// (tensor_load_to_lds) DMAs the dry window + 4 tap windows into LDS, compute
// reads b128 from LDS, stores b128 to global.
// ---------------------------------------------------------------------------

#define TILE 2048          // fp32 elements per block tile
#define NWIN 5             // dry + 4 taps
#define BLOCK 256          // 8 waves (wave32)

__device__ __constant__ const int   kShift[4] = {11026, 22051, 33076, 44101};
#define S_MAX 44101
#define G0 0.8f
#define G1 0.64f
#define G2 0.512f
#define G3 0.4096f

typedef unsigned int uint_t;
typedef uint_t v4u __attribute__((ext_vector_type(4)));
typedef int    v4i __attribute__((ext_vector_type(4)));
typedef int    v8i __attribute__((ext_vector_type(8)));

#if defined(__has_builtin)
#if __has_builtin(__builtin_amdgcn_tensor_load_to_lds)
#define HAVE_TDM 1
#endif
#endif

#ifdef HAVE_TDM
// Issue one TDM load of a 1-row tile: tile_elems fp32 from gptr -> LDS offset
// lds_byte. Elements with X >= tensor_elems are zero-filled by TDM OOB rules.
__device__ __forceinline__ void tdm_load_1d(uint_t lds_byte, const float* gptr,
                                            uint_t tile_elems, uint_t tensor_elems) {
  uint64_t ga = (uint64_t)(uintptr_t)gptr;

  // D# group 0 (128b): count=1 | lds_addr | global_addr[56:0] | type=2
  v4u g0;
  g0.x = 1u;                                        // count=1, user mode, no gather
  g0.y = lds_byte;                                  // lds_addr (bytes)
  g0.z = (uint_t)ga;                                // global_addr[31:0]
  g0.w = ((uint_t)(ga >> 32) & 0x01FFFFFFu) | (2u << 30);  // addr[56:32], type=2

  // D# group 1 (256b): data_size=4B, tensor_dim0, tensor_dim1=1,
  // tile_dim0, tile_dim1=1, tensor_dim0_stride
  uint64_t stride0 = (uint64_t)tensor_elems;
  v8i g1;
  g1[0] = (int)(2u << 16);                                   // data_size = 4B
  g1[1] = (int)((tensor_elems & 0xFFFFu) << 16);             // tensor_dim0[15:0]
  g1[2] = (int)((tensor_elems >> 16) | (1u << 16));          // tdim0[31:16] | tdim1 lo=1
  g1[3] = (int)(tile_elems << 16);                           // tdim1 hi=0 | tile_dim0
  g1[4] = (int)1u;                                           // tile_dim1=1, tile_dim2=0
  g1[5] = (int)(uint_t)(stride0 & 0xFFFFFFFFu);              // dim0_stride[31:0]
  g1[6] = (int)(uint_t)((stride0 >> 32) & 0xFFFFu);          // dim0_stride[47:32]
  g1[7] = 0;

  v4i z4 = {0, 0, 0, 0};
#if __clang_major__ >= 23
  v8i z8 = {0, 0, 0, 0, 0, 0, 0, 0};
  __builtin_amdgcn_tensor_load_to_lds(g0, g1, z4, z4, z8, 0);
#else
  __builtin_amdgcn_tensor_load_to_lds(g0, g1, z4, z4, 0);
#endif
}
#endif

__global__ __launch_bounds__(BLOCK)
void MultiTapDelay_kernel(const float* __restrict__ x, float* __restrict__ out,
                          long long n) {
  extern __shared__ float smem[];  // NWIN * TILE floats

  const long long tileStart = (long long)blockIdx.x * TILE;
  if (tileStart >= n) return;
  const bool steady = (tileStart >= (long long)S_MAX);
  const long long rem = n - tileStart;
  const uint_t dryLen = rem < (long long)TILE ? (uint_t)rem : (uint_t)TILE;

#ifdef HAVE_TDM
  if ((threadIdx.x >> 5) == 0) {  // wave 0 issues the DMAs (TDM ignores EXEC)
    tdm_load_1d(0u, x + tileStart, TILE, dryLen);  // dry; zero-fill past n
    if (steady) {
      tdm_load_1d(1u * TILE * 4u, x + (tileStart - kShift[0]), TILE, TILE);
      tdm_load_1d(2u * TILE * 4u, x + (tileStart - kShift[1]), TILE, TILE);
      tdm_load_1d(3u * TILE * 4u, x + (tileStart - kShift[2]), TILE, TILE);
      tdm_load_1d(4u * TILE * 4u, x + (tileStart - kShift[3]), TILE, TILE);
    }
#if __has_builtin(__builtin_amdgcn_s_wait_tensorcnt)
    __builtin_amdgcn_s_wait_tensorcnt(0);
#else
    asm volatile("s_wait_tensorcnt 0x0" ::: "memory");
#endif
  }
  asm volatile("" ::: "memory");
#else
  // Fallback: fill LDS with plain coalesced loads.
  for (int i = threadIdx.x; i < TILE; i += blockDim.x) {
    long long t = tileStart + i;
    smem[i] = (t < n) ? x[t] : 0.0f;
    if (steady) {
      smem[1 * TILE + i] = x[t - kShift[0]];
      smem[2 * TILE + i] = x[t - kShift[1]];
      smem[3 * TILE + i] = x[t - kShift[2]];
      smem[4 * TILE + i] = x[t - kShift[3]];
    }
  }
#endif
  __syncthreads();

  for (int i = (int)threadIdx.x * 4; i < TILE; i += (int)blockDim.x * 4) {
    long long t = tileStart + i;
    if (t >= n) break;

    const float4 dry = *reinterpret_cast<const float4*>(&smem[i]);
    float4 wet;
    if (steady) {
      const float4 w0 = *reinterpret_cast<const float4*>(&smem[1 * TILE + i]);
      const float4 w1 = *reinterpret_cast<const float4*>(&smem[2 * TILE + i]);
      const float4 w2 = *reinterpret_cast<const float4*>(&smem[3 * TILE + i]);
      const float4 w3 = *reinterpret_cast<const float4*>(&smem[4 * TILE + i]);
      wet.x = G0 * w0.x + G1 * w1.x + G2 * w2.x + G3 * w3.x;
      wet.y = G0 * w0.y + G1 * w1.y + G2 * w2.y + G3 * w3.y;
      wet.z = G0 * w0.z + G1 * w1.z + G2 * w2.z + G3 * w3.z;
      wet.w = G0 * w0.w + G1 * w1.w + G2 * w2.w + G3 * w3.w;
    } else {
      // Head region (22 of 12920 blocks): predicated global tap reads.
      float acc[4];
#pragma unroll
      for (int j = 0; j < 4; ++j) {
        long long tj = t + j;
        float a = 0.0f;
        if (tj >= kShift[0]) a += G0 * x[tj - kShift[0]];
        if (tj >= kShift[1]) a += G1 * x[tj - kShift[1]];
        if (tj >= kShift[2]) a += G2 * x[tj - kShift[2]];
        if (tj >= kShift[3]) a += G3 * x[tj - kShift[3]];
        acc[j] = a;
      }
      wet.x = acc[0]; wet.y = acc[1]; wet.z = acc[2]; wet.w = acc[3];
    }

    float4 o;
    o.x = 0.5f * dry.x + 0.5f * wet.x;
    o.y = 0.5f * dry.y + 0.5f * wet.y;
    o.z = 0.5f * dry.z + 0.5f * wet.z;
    o.w = 0.5f * dry.w + 0.5f * wet.w;

    if (t + 3 < n) {
      *reinterpret_cast<float4*>(&out[t]) = o;  // global_store_b128
    } else {
#pragma unroll
      for (int j = 0; j < 4; ++j)
        if (t + j < n) out[t + j] = (&o.x)[j];
    }
  }
}

extern "C" void kernel_launch(void* const* d_in, const int* in_sizes, int n_in,
                              void* d_out, int out_size, void* d_ws, size_t ws_size,
                              hipStream_t stream) {
  (void)n_in; (void)d_ws; (void)ws_size; (void)out_size;
  const float* x = (const float*)d_in[0];
  float* out = (float*)d_out;
  const long long n = (long long)in_sizes[0];
  const int nBlocks = (int)((n + TILE - 1) / TILE);
  const size_t shbytes = (size_t)NWIN * TILE * sizeof(float);
  MultiTapDelay_kernel<<<nBlocks, BLOCK, shbytes, stream>>>(x, out, n);
}